// SAGEConv_25950192402572
// MI455X (gfx1250) — compile-verified
//
#include <hip/hip_runtime.h>

#define N_NODES 100000
#define N_EDGES 640000
#define D 128

typedef float v2f __attribute__((ext_vector_type(2)));
typedef float v8f __attribute__((ext_vector_type(8)));

// ---------------------------------------------------------------------------
// Kernel 1: zero the workspace accumulators (neigh_sum[N*D] + deg[N]).
// ---------------------------------------------------------------------------
__global__ void sage_zero_ws(float* __restrict__ ws, int n) {
    int i = blockIdx.x * blockDim.x + threadIdx.x;
    if (i < n) ws[i] = 0.0f;
}

// ---------------------------------------------------------------------------
// Kernel 2: edge scatter.  32 lanes per edge; each lane moves a float4 of the
// source node's features into the destination accumulator with hardware FP32
// atomics (L2-resident: 51.2 MB feature table << 192 MB L2).
// ---------------------------------------------------------------------------
__global__ void sage_edge_scatter(const float* __restrict__ feat,
                                  const int*   __restrict__ src,
                                  const int*   __restrict__ dst,
                                  float*       __restrict__ neigh_sum,
                                  float*       __restrict__ deg) {
    int tid = blockIdx.x * blockDim.x + threadIdx.x;
    int e = tid >> 5;          // edge index
    int c = tid & 31;          // 32 chunks of 4 floats = 128 features
    if (e >= N_EDGES) return;
    int s  = src[e];
    int dn = dst[e];
    const float4 v = *(const float4*)(feat + (size_t)s * D + c * 4);
    float* o = neigh_sum + (size_t)dn * D + c * 4;
    unsafeAtomicAdd(o + 0, v.x);   // -> global_atomic_add_f32
    unsafeAtomicAdd(o + 1, v.y);
    unsafeAtomicAdd(o + 2, v.z);
    unsafeAtomicAdd(o + 3, v.w);
    if (c == 0) unsafeAtomicAdd(deg + dn, 1.0f);
}

// ---------------------------------------------------------------------------
// Kernel 3: fused dual GEMM via V_WMMA_F32_16X16X4_F32 (exact fp32 path).
//   out[16r : 16r+16, :] = feat_tile @ W_self + (neigh_sum_tile * rdeg) @ W_neigh
//                          + (b_self + b_neigh)
// One wave per 16x16 output tile; 8 waves/block cover all 128 columns.
// 6250 blocks * 16 rows == 100000 exactly -> no divergence (EXEC all 1s).
// ---------------------------------------------------------------------------
__global__ __launch_bounds__(256)
void sage_wmma_gemm(const float* __restrict__ feat,
                    const float* __restrict__ neigh_sum,
                    const float* __restrict__ deg,
                    const float* __restrict__ Wself,
                    const float* __restrict__ bself,
                    const float* __restrict__ Wneigh,
                    const float* __restrict__ bneigh,
                    float*       __restrict__ out) {
    const int lane  = threadIdx.x & 31;
    const int wave  = threadIdx.x >> 5;       // 0..7 -> column tile
    const int rloc  = lane & 15;              // row within 16-row tile (A/D)
    const int khalf = lane >> 4;              // 0: K pair {0,1}, 1: K pair {2,3}
    const int row   = blockIdx.x * 16 + rloc;
    const int col   = wave * 16 + (lane & 15);

    // Accumulator starts at the (column-only) bias; same value in all 8 VGPRs
    // since D-matrix VGPR j differs only in row.
    const float bias = bself[col] + bneigh[col];
    v8f acc;
#pragma unroll
    for (int j = 0; j < 8; ++j) acc[j] = bias;

    const float* frow = feat      + (size_t)row * D;
    const float* nrow = neigh_sum + (size_t)row * D;
    const float  rdeg = 1.0f / fmaxf(deg[row], 1.0f);

    // --- self term: feat @ W_self, K = 128 in steps of 4 -------------------
#pragma unroll
    for (int k0 = 0; k0 < D; k0 += 4) {
        const int k = k0 + khalf * 2;
        v2f a; a.x = frow[k];  a.y = frow[k + 1];
        v2f b; b.x = Wself[(size_t)k * D + col];
               b.y = Wself[(size_t)(k + 1) * D + col];
        acc = __builtin_amdgcn_wmma_f32_16x16x4_f32(
                  false, a, false, b, (short)0, acc, false, false);
    }

    // --- neighbor term: (neigh_sum * rdeg) @ W_neigh -----------------------
#pragma unroll
    for (int k0 = 0; k0 < D; k0 += 4) {
        const int k = k0 + khalf * 2;
        v2f a; a.x = nrow[k] * rdeg;  a.y = nrow[k + 1] * rdeg;
        v2f b; b.x = Wneigh[(size_t)k * D + col];
               b.y = Wneigh[(size_t)(k + 1) * D + col];
        acc = __builtin_amdgcn_wmma_f32_16x16x4_f32(
                  false, a, false, b, (short)0, acc, false, false);
    }

    // --- store D: VGPR j holds row (j + khalf*8), col = lane&15 ------------
    float* obase = out + (size_t)(blockIdx.x * 16 + khalf * 8) * D + col;
#pragma unroll
    for (int j = 0; j < 8; ++j)
        obase[(size_t)j * D] = acc[j];
}

// ---------------------------------------------------------------------------
extern "C" void kernel_launch(void* const* d_in, const int* in_sizes, int n_in,
                              void* d_out, int out_size, void* d_ws, size_t ws_size,
                              hipStream_t stream) {
    const float* feat   = (const float*)d_in[0];
    const int*   src    = (const int*)  d_in[1];
    const int*   dst    = (const int*)  d_in[2];
    const float* Wself  = (const float*)d_in[3];
    const float* bself  = (const float*)d_in[4];
    const float* Wneigh = (const float*)d_in[5];
    const float* bneigh = (const float*)d_in[6];
    float*       out    = (float*)d_out;

    float* neigh_sum = (float*)d_ws;                    // N_NODES * D floats
    float* deg       = neigh_sum + (size_t)N_NODES * D; // N_NODES floats

    const int zn = N_NODES * (D + 1);
    sage_zero_ws<<<(zn + 255) / 256, 256, 0, stream>>>((float*)d_ws, zn);

    const int et = N_EDGES * 32;
    sage_edge_scatter<<<(et + 255) / 256, 256, 0, stream>>>(
        feat, src, dst, neigh_sum, deg);

    sage_wmma_gemm<<<N_NODES / 16, 256, 0, stream>>>(
        feat, neigh_sum, deg, Wself, bself, Wneigh, bneigh, out);
}